// Criterion_67250597921404
// MI455X (gfx1250) — compile-verified
//
#include <hip/hip_runtime.h>

// Clang extended vectors (HIP's float4 is a struct; ext vectors are what the
// WMMA builtins and nontemporal-load builtins want).
typedef float v2f __attribute__((ext_vector_type(2)));
typedef float v4f __attribute__((ext_vector_type(4)));
typedef float v8f __attribute__((ext_vector_type(8)));

// ---------------------------------------------------------------------------
// Wave32 cross-lane sum via V_WMMA_F32_16X16X4_F32.
//   A (16x4 f32): lane L<16 holds {K0,K1} = {v,0}; lane L>=16 holds {K2,K3} =
//   {v,0} for row M = L%16.  B = all-ones (4x16).
//   => D[m][n] = v[m] + v[m+16] for every n.
//   VGPR r of D holds rows m=r (lanes 0-15) and m=r+8 (lanes 16-31), so
//   sum(D[0..7]) per lane covers rows 0-7 (lo half) / 8-15 (hi half);
//   one shfl_xor(16) completes the full 32-lane sum in every lane.
// EXEC must be all ones: callers invoke this unconditionally from full waves.
// ---------------------------------------------------------------------------
__device__ __forceinline__ float wave_sum_wmma(float v) {
    v2f a; a.x = v;    a.y = 0.0f;
    v2f b; b.x = 1.0f; b.y = 1.0f;
    v8f c = {};
    c = __builtin_amdgcn_wmma_f32_16x16x4_f32(
        /*neg_a=*/false, a, /*neg_b=*/false, b,
        /*c_mod=*/(short)0, c, /*reuse_a=*/false, /*reuse_b=*/false);
    float s = ((c[0] + c[1]) + (c[2] + c[3])) + ((c[4] + c[5]) + (c[6] + c[7]));
    s += __shfl_xor(s, 16, 32);   // pair the two row-halves -> full wave sum
    return s;
}

// Block reduction: WMMA per wave, then 8 wave results combined through LDS.
// Valid result only in threadIdx.x == 0. Block size must be 256 (8 wave32s).
__device__ __forceinline__ float block_sum(float v) {
    __shared__ float wsum[8];
    float s = wave_sum_wmma(v);               // all lanes of wave hold wave sum
    const int lane = threadIdx.x & 31;
    const int wave = threadIdx.x >> 5;
    if (lane == 0) wsum[wave] = s;
    __syncthreads();
    float t = 0.0f;
    if (threadIdx.x == 0) {
        #pragma unroll
        for (int i = 0; i < 8; ++i) t += wsum[i];
    }
    return t;
}

// ---------------------------------------------------------------------------
// Pass 1: streaming grid-stride over float4 triples; per-block partial sums.
// e = (t-mu)^2/sigma + log(sigma), accumulated per-lane, reduced at the end.
// ---------------------------------------------------------------------------
__global__ __launch_bounds__(256)
void gnll_partial(const float* __restrict__ mu,
                  const float* __restrict__ sg,
                  const float* __restrict__ tg,
                  float* __restrict__ partial,
                  long long n4) {
    const v4f* __restrict__ m4 = (const v4f*)mu;
    const v4f* __restrict__ s4 = (const v4f*)sg;
    const v4f* __restrict__ t4 = (const v4f*)tg;

    long long i      = (long long)blockIdx.x * blockDim.x + threadIdx.x;
    long long stride = (long long)gridDim.x * blockDim.x;

    float acc = 0.0f;
    for (; i < n4; i += stride) {
        v4f m = __builtin_nontemporal_load(m4 + i);   // global_load_b128, TH=NT
        v4f s = __builtin_nontemporal_load(s4 + i);
        v4f t = __builtin_nontemporal_load(t4 + i);
        v4f d = t - m;
        float e0 = __fdividef(d.x * d.x, s.x) + __logf(s.x);
        float e1 = __fdividef(d.y * d.y, s.y) + __logf(s.y);
        float e2 = __fdividef(d.z * d.z, s.z) + __logf(s.z);
        float e3 = __fdividef(d.w * d.w, s.w) + __logf(s.w);
        acc += (e0 + e1) + (e2 + e3);
    }

    float total = block_sum(acc);                     // uniform control flow
    if (threadIdx.x == 0) partial[blockIdx.x] = total;
}

// ---------------------------------------------------------------------------
// Pass 2: single block folds the block partials and applies the affine terms.
// loss = 0.5 * total / B + 0.5 * D * log(2*pi)
// ---------------------------------------------------------------------------
__global__ __launch_bounds__(256)
void gnll_final(const float* __restrict__ partial, int n,
                float* __restrict__ out, long long B, int D) {
    float acc = 0.0f;
    for (int i = threadIdx.x; i < n; i += 256) acc += partial[i];
    float total = block_sum(acc);
    if (threadIdx.x == 0) {
        const double LOG_2PI = 1.8378770664093453;    // log(2*pi)
        double r = 0.5 * (double)total / (double)B + 0.5 * (double)D * LOG_2PI;
        out[0] = (float)r;
    }
}

extern "C" void kernel_launch(void* const* d_in, const int* in_sizes, int n_in,
                              void* d_out, int out_size, void* d_ws, size_t ws_size,
                              hipStream_t stream) {
    const float* mu = (const float*)d_in[0];
    const float* sg = (const float*)d_in[1];
    const float* tg = (const float*)d_in[2];
    float* out      = (float*)d_out;
    float* partial  = (float*)d_ws;

    const long long N = (long long)in_sizes[0];       // B * D
    const int       D = 2048;                          // per reference
    const long long B = (D > 0 && N % D == 0) ? N / D : N;
    const long long n4 = N / 4;                        // D % 4 == 0

    int blocks = 2048;                                 // 8.4M float4s / 512K lanes = 16 iters
    if ((size_t)blocks * sizeof(float) > ws_size)
        blocks = (int)(ws_size / sizeof(float));
    if (blocks < 1) blocks = 1;

    gnll_partial<<<blocks, 256, 0, stream>>>(mu, sg, tg, partial, n4);
    gnll_final  <<<1,      256, 0, stream>>>(partial, blocks, out, B, D);
}